// ReversiSmallModel_78554951844377
// MI455X (gfx1250) — compile-verified
//
#include <hip/hip_runtime.h>
#include <hip/hip_bf16.h>

typedef __attribute__((ext_vector_type(2))) float v2f;
typedef __attribute__((ext_vector_type(8))) float v8f;

#define NFEATS      32
#define GROUP_SIZE  8192
#define VOCAB       (NFEATS * GROUP_SIZE)   // 262144
#define LPA         128
#define BATCH       16384
#define NHEADS      30

// ---------------------------------------------------------------------------
// Kernel 1: phase-adaptive gather + sum + bias + clip  ->  x[B,128] (f32, ws)
// 2 samples per 256-thread block; thread owns one of 128 dims.
// Fully unrolled 32-row gather => 32 loads in flight per thread (memory-bound,
// random 512B rows, coalesced across the 128 dim-lanes).
// ---------------------------------------------------------------------------
__global__ void __launch_bounds__(256)
gather_act_kernel(const int* __restrict__ fidx,     // [B,32]
                  const int* __restrict__ ply,      // [B]
                  const int* __restrict__ foff,     // [32]
                  const float* __restrict__ emb,    // [3*VOCAB,128]
                  const float* __restrict__ pa_bias,// [3,128]
                  float* __restrict__ x)            // [B,128] out
{
    __shared__ int sidx[2][NFEATS];
    const int local = threadIdx.x & 127;            // dim 0..127
    const int sloc  = threadIdx.x >> 7;             // 0..1
    const int s     = blockIdx.x * 2 + sloc;        // sample

    if (local < NFEATS) {
        sidx[sloc][local] = fidx[s * NFEATS + local] + foff[local];
    }
    __syncthreads();

    const int pa = ply[s] / 10;                     // phase bucket 0..2
    const float* __restrict__ base =
        emb + (size_t)pa * ((size_t)VOCAB * LPA);

    float acc = 0.0f;
#pragma unroll
    for (int f = 0; f < NFEATS; ++f) {
        const float* row = base + (size_t)sidx[sloc][f] * LPA;
        acc += row[local];
    }
    acc += pa_bias[pa * LPA + local];
    acc = fminf(fmaxf(acc, 0.0f), 1.0f);            // NNUE clipped activation
    x[(size_t)s * LPA + local] = acc;
}

// ---------------------------------------------------------------------------
// Kernel 2: y = x(16x128) @ w_out^T(128x30 padded to 32) per 16-sample tile
// using V_WMMA_F32_16X16X4_F32 (full f32 precision), then per-sample head
// selection via LDS. One wave per tile, 8 waves per block.
// ---------------------------------------------------------------------------
__global__ void __launch_bounds__(256)
heads_wmma_kernel(const float* __restrict__ x,      // [B,128]
                  const int* __restrict__ ply,      // [B]
                  const float* __restrict__ w_out,  // [30,128]
                  const float* __restrict__ b_out,  // [30]
                  float* __restrict__ out)          // [B]
{
    __shared__ float ytile[8][16 * 32];

    const int lane = threadIdx.x & 31;
    const int wave = threadIdx.x >> 5;
    const int tile = blockIdx.x * 8 + wave;         // 16-sample tile id
    const int m    = lane & 15;                     // row (A) / col (B) index
    const int khi  = lane >> 4;                     // K-pair select: {0,1} / {2,3}

    // A fragment source: sample row of x (each lane covers half its K values)
    const float* xrow  = x + (size_t)(tile * 16 + m) * LPA + khi * 2;
    // B fragments: B[k][n] = w_out[n][k]. half0: n = m (0..15)
    const float* w0row = w_out + m * LPA + khi * 2;
    // half1: n = 16+m (rows 30,31 are padding -> clamp index, mask to zero)
    const int   n1    = 16 + m;
    const float msk   = (n1 < NHEADS) ? 1.0f : 0.0f;
    const int   n1c   = (n1 < NHEADS) ? n1 : 0;
    const float* w1row = w_out + n1c * LPA + khi * 2;

    v8f c0 = {};
    v8f c1 = {};
#pragma unroll 8
    for (int kk = 0; kk < 32; ++kk) {
        const int o = kk * 4;
        v2f a  = *(const v2f*)(xrow  + o);
        v2f b0 = *(const v2f*)(w0row + o);
        v2f b1 = *(const v2f*)(w1row + o);
        b1 *= msk;
        c0 = __builtin_amdgcn_wmma_f32_16x16x4_f32(
                 false, a, false, b0, (short)0, c0, false, false);
        c1 = __builtin_amdgcn_wmma_f32_16x16x4_f32(
                 false, a, false, b1, (short)0, c1, false, false);
    }

    // Spill the 16x32 result tile to LDS (C/D layout: VGPR v, lane ->
    // row = v + 8*khi, col = m (+16 for the second tile))
    float* yl = &ytile[wave][0];
#pragma unroll
    for (int v = 0; v < 8; ++v) {
        const int row = v + 8 * khi;
        yl[row * 32 + m]      = c0[v];
        yl[row * 32 + 16 + m] = c1[v];
    }
    __syncthreads();

    // Per-sample head selection: ls = ply // 1 in [0,30)
    if (lane < 16) {
        const int b  = tile * 16 + lane;
        const int ls = ply[b];
        out[b] = yl[lane * 32 + ls] + b_out[ls];
    }
}

// ---------------------------------------------------------------------------
extern "C" void kernel_launch(void* const* d_in, const int* in_sizes, int n_in,
                              void* d_out, int out_size, void* d_ws, size_t ws_size,
                              hipStream_t stream) {
    const int*   fidx    = (const int*)d_in[0];    // feature_indices [B,32]
    const int*   ply     = (const int*)d_in[1];    // ply [B]
    const int*   foff    = (const int*)d_in[2];    // feature_offsets [32]
    const float* emb     = (const float*)d_in[3];  // [3*VOCAB,128]
    const float* pa_bias = (const float*)d_in[4];  // [3,128]
    const float* w_out   = (const float*)d_in[5];  // [30,128]
    const float* b_out   = (const float*)d_in[6];  // [30]
    float*       out     = (float*)d_out;          // [B,1]
    float*       xbuf    = (float*)d_ws;           // [B,128] activations (8 MB)

    gather_act_kernel<<<BATCH / 2, 256, 0, stream>>>(fidx, ply, foff, emb,
                                                     pa_bias, xbuf);

    // 1024 tiles of 16 samples, 8 waves (tiles) per block -> 128 blocks
    heads_wmma_kernel<<<(BATCH / 16) / 8, 256, 0, stream>>>(xbuf, ply, w_out,
                                                            b_out, out);
}